// DigitCaps_1030792151656
// MI455X (gfx1250) — compile-verified
//
#include <hip/hip_runtime.h>
#include <hip/hip_bf16.h>

// CDNA5 (gfx1250) fused CapsNet dynamic-routing kernel.
//
// Key idea: x_hat[b,o,i,:] = W[o,i] (16x8) @ x[b,i,:] is never materialized
// (377 MB). Routing logits are recomputable as dot(vsum, x_hat) where
// vsum = sum of squashed outputs of previous iterations, so the whole
// 3-iteration routing loop fuses into one kernel with only LDS state.
//
// WMMA mapping (V_WMMA_F32_16X16X4_F32, two K=4 chunks for D_in=8):
//   D[d, b] = sum_e A[d,e] * B[e,b],  A = W[o,i] tile, B = x[btile,i] tile
// C/D layout: VGPR r, lane l -> d = r + 8*(l>=16), b = l&15  (batch on N!)
//   -> per-batch scalars are lane-local; c[b,o,i] folds into the B operand
//      so s[b,o,:] += c * x_hat accumulates inside the WMMA C operand.

typedef __attribute__((ext_vector_type(2))) float v2f;
typedef __attribute__((ext_vector_type(8))) float v8f;

#define B_TOT    512
#define O_CAPS   10
#define I_CAPS   1152
#define D_OUT    16
#define D_IN     8
#define BT       16          // batch tile = WMMA N
#define NTHREADS 256         // 8 wave32 waves
#define NWAVES   (NTHREADS / 32)
#define N_ITERS  3

__global__ __launch_bounds__(NTHREADS)
void digitcaps_fused(const float* __restrict__ x,
                     const float* __restrict__ Wt,
                     float* __restrict__ out)
{
    __shared__ float vsum_lds[O_CAPS * BT * D_OUT]; // sum of v over past iters
    __shared__ float s_lds[O_CAPS * BT * D_OUT];    // s, then v (in place)
    __shared__ float sn_lds[O_CAPS * BT];           // ||s||^2 (monotone w/ ||v||)

    const int tid  = threadIdx.x;
    const int lane = tid & 31;
    const int wave = tid >> 5;
    const int hi   = lane >> 4;    // 0/1: which K/D half this lane holds
    const int ln   = lane & 15;    // A rows: d index; B/C cols: batch index
    const int b0   = blockIdx.x * BT;

    for (int t = tid; t < O_CAPS * BT * D_OUT; t += NTHREADS) vsum_lds[t] = 0.f;
    __syncthreads();

    for (int it = 0; it < N_ITERS; ++it) {
        for (int t = tid; t < O_CAPS * BT * D_OUT; t += NTHREADS) s_lds[t] = 0.f;
        __syncthreads();

        // per-wave partial s accumulators (live in WMMA C registers)
        v8f sacc[O_CAPS];
#pragma unroll
        for (int o = 0; o < O_CAPS; ++o) sacc[o] = (v8f){};

        for (int i = wave; i < I_CAPS; i += NWAVES) {
            // B operand = x[b0+ln, i, :]^T (K=e, N=b). 32-bit B 4x16 layout:
            // VGPR0 -> K = 2*hi, VGPR1 -> K = 2*hi+1, column = ln.
            const float* xp = x + ((size_t)(b0 + ln) * I_CAPS + i) * D_IN;
            const v2f bx0 = *(const v2f*)(xp + 2 * hi);       // e-chunk 0..3
            const v2f bx1 = *(const v2f*)(xp + 4 + 2 * hi);   // e-chunk 4..7

            v2f aw0[O_CAPS], aw1[O_CAPS];
            float logit[O_CAPS];
#pragma unroll
            for (int o = 0; o < O_CAPS; ++o) {
                // A operand = W[o,i,d=ln, e]: VGPR0 -> K=2*hi, VGPR1 -> K=2*hi+1
                const float* wp =
                    Wt + (((size_t)o * I_CAPS + i) * D_OUT + ln) * D_IN;
                aw0[o] = *(const v2f*)(wp + 2 * hi);
                aw1[o] = *(const v2f*)(wp + 4 + 2 * hi);

                v8f xh = {};
                xh = __builtin_amdgcn_wmma_f32_16x16x4_f32(
                        false, aw0[o], false, bx0, (short)0, xh, false, false);
                xh = __builtin_amdgcn_wmma_f32_16x16x4_f32(
                        false, aw1[o], false, bx1, (short)0, xh, false, false);

                // routing logit: dot_d(vsum[b,o,:], x_hat[b,o,i,:])
                // lane holds d = r + 8*hi for b = ln -> 8 FMAs + 1 shfl_xor(16)
                const float* vsp = &vsum_lds[(o * BT + ln) * D_OUT + 8 * hi];
                float p = 0.f;
#pragma unroll
                for (int r = 0; r < 8; ++r) p += vsp[r] * xh[r];
                p += __shfl_xor(p, 16, 32);
                logit[o] = p;
            }

            // softmax over output capsules (per lane; this lane's batch = ln)
            float m = logit[0];
#pragma unroll
            for (int o = 1; o < O_CAPS; ++o) m = fmaxf(m, logit[o]);
            float ce[O_CAPS];
            float den = 0.f;
#pragma unroll
            for (int o = 0; o < O_CAPS; ++o) {
                ce[o] = __expf(logit[o] - m);
                den += ce[o];
            }
            const float inv = 1.f / den;

            // s[b,o,:] += c[b,o,i] * x_hat[b,o,i,:]
            // c is a per-column (per-batch) scale -> fold into B operand and
            // let the WMMA C accumulator do the reduction over i.
#pragma unroll
            for (int o = 0; o < O_CAPS; ++o) {
                const float co = ce[o] * inv;
                const v2f cb0 = bx0 * co;
                const v2f cb1 = bx1 * co;
                sacc[o] = __builtin_amdgcn_wmma_f32_16x16x4_f32(
                        false, aw0[o], false, cb0, (short)0, sacc[o], false, false);
                sacc[o] = __builtin_amdgcn_wmma_f32_16x16x4_f32(
                        false, aw1[o], false, cb1, (short)0, sacc[o], false, false);
            }
        }

        // cross-wave reduction of s partials (ds_add_f32)
#pragma unroll
        for (int o = 0; o < O_CAPS; ++o) {
#pragma unroll
            for (int r = 0; r < 8; ++r) {
                atomicAdd(&s_lds[(o * BT + ln) * D_OUT + (r + 8 * hi)], sacc[o][r]);
            }
        }
        __syncthreads();

        // squash: v = s * sqrt(sn)/(1+sn); vsum += v; keep sn for argmax
        if (tid < O_CAPS * BT) {
            const int o  = tid / BT;
            const int bb = tid % BT;
            float* sp = &s_lds[(o * BT + bb) * D_OUT];
            float sv[D_OUT];
            float sn = 0.f;
#pragma unroll
            for (int d = 0; d < D_OUT; ++d) { sv[d] = sp[d]; sn += sv[d] * sv[d]; }
            const float scale = (sn > 0.f) ? (sqrtf(sn) / (1.f + sn)) : 0.f;
            float* vp = &vsum_lds[(o * BT + bb) * D_OUT];
#pragma unroll
            for (int d = 0; d < D_OUT; ++d) {
                const float v = sv[d] * scale;
                sp[d] = v;          // s_lds now holds v
                vp[d] += v;
            }
            sn_lds[o * BT + bb] = sn;   // ||v|| is monotone in sn
        }
        __syncthreads();
    }

    // outputs [B, O, 16] live at flat offset B*16 (tuple order: t, outputs)
    for (int t = tid; t < O_CAPS * BT * D_OUT; t += NTHREADS) {
        const int o  = t / (BT * D_OUT);
        const int bb = (t / D_OUT) % BT;
        const int d  = t % D_OUT;
        out[(size_t)B_TOT * D_OUT +
            (((size_t)(b0 + bb)) * O_CAPS + o) * D_OUT + d] =
            s_lds[(o * BT + bb) * D_OUT + d];
    }
    // t [B, 1, 16]: pick longest capsule (argmax over sn == argmax of softmax)
    if (tid < BT * D_OUT) {
        const int bb = tid / D_OUT;
        const int d  = tid % D_OUT;
        int   best = 0;
        float bm   = sn_lds[bb];
        for (int o = 1; o < O_CAPS; ++o) {
            const float v = sn_lds[o * BT + bb];
            if (v > bm) { bm = v; best = o; }
        }
        out[((size_t)(b0 + bb)) * D_OUT + d] = s_lds[(best * BT + bb) * D_OUT + d];
    }
}

extern "C" void kernel_launch(void* const* d_in, const int* in_sizes, int n_in,
                              void* d_out, int out_size, void* d_ws, size_t ws_size,
                              hipStream_t stream) {
    (void)in_sizes; (void)n_in; (void)d_ws; (void)ws_size; (void)out_size;
    const float* x = (const float*)d_in[0];   // [512, 1152, 8]
    const float* W = (const float*)d_in[1];   // [10, 1152, 16, 8]
    float* out = (float*)d_out;               // [512*16] ++ [512*10*16]
    digitcaps_fused<<<dim3(B_TOT / BT), dim3(NTHREADS), 0, stream>>>(x, W, out);
}